// WinDecoder_69286412419395
// MI455X (gfx1250) — compile-verified
//
#include <hip/hip_runtime.h>

// ---------------------------------------------------------------------------
// WinDecoder on MI455X (gfx1250).
// Heavy math: 496-pair MLP scorer = 496x3136 @ 3136x256 GEMM (+relu, +dot W2).
// ~0.8 GFLOP / ~10MB -> latency bound; use fp32 WMMA (v_wmma_f32_16x16x4_f32)
// to keep reference fp32 precision while issuing on the matrix pipes.
// Beam search (sequential, tiny state) runs in one persistent 128-thread block.
// ---------------------------------------------------------------------------

typedef __attribute__((ext_vector_type(2))) float v2f;
typedef __attribute__((ext_vector_type(8))) float v8f;

#define WD_NEG_INF (-1.0e30f)

constexpr int N_M   = 32;     // mentions
constexpr int D     = 1024;   // mention dim
constexpr int PDIM  = 64;     // pair-embedding dim
constexpr int INF_K = 3*D + PDIM;  // 3136
constexpr int HID   = 256;
constexpr int NPAIR = 496;    // all (a,b), a<b
constexpr int MPAD  = 512;    // pad M to multiple of 16

__device__ __forceinline__ int pair_off(int w) {
  // #pairs with gap < w :  sum_{j=1..w-1} (32 - j)
  return (w - 1) * N_M - ((w - 1) * w) / 2;
}

// ---------------------------------------------------------------------------
// K0: build feats F[p] = [m_a, m_b, m_a*m_b, dist_emb[w]]   (pad rows -> 0)
// ---------------------------------------------------------------------------
__global__ __launch_bounds__(256) void wd_build_feats(
    const float* __restrict__ mentions, const float* __restrict__ dist_emb,
    float* __restrict__ F)
{
  const int p = blockIdx.x;
  float* row = F + (size_t)p * INF_K;
  if (p >= NPAIR) {
    for (int c = threadIdx.x; c < INF_K; c += 256) row[c] = 0.0f;
    return;
  }
  int rem = p, w = 1;
  while (rem >= N_M - w) { rem -= N_M - w; ++w; }
  const int a = rem;
  const int b = rem + w;
  const float* ma = mentions + a * D;
  const float* mb = mentions + b * D;
  const float* de = dist_emb + w * PDIM;   // clip(b-a,0,63) == w here
  for (int c = threadIdx.x; c < INF_K; c += 256) {
    float v;
    if      (c < D)     v = ma[c];
    else if (c < 2*D)   v = mb[c - D];
    else if (c < 3*D)   v = ma[c - 2*D] * mb[c - 2*D];
    else                v = de[c - 3*D];
    row[c] = v;
  }
}

// ---------------------------------------------------------------------------
// K1: H = relu(F @ W1 + b1), fp32 WMMA 16x16x4. One wave per 16x16 tile.
//   A frag (16x4):  lane L holds row M=L%16, VGPR j holds K = 2*(L/16)+j
//   B frag (4x16):  lane L holds col N=L%16, VGPR j holds K = 2*(L/16)+j
//   C/D (16x16):    VGPR j, lane L -> M = j + 8*(L/16), N = L%16
// ---------------------------------------------------------------------------
__global__ __launch_bounds__(32) void wd_gemm_relu(
    const float* __restrict__ F, const float* __restrict__ W1,
    const float* __restrict__ b1, float* __restrict__ H)
{
  const int lane = threadIdx.x;
  const int l16  = lane & 15;
  const int half = lane >> 4;
  const int mt = blockIdx.x;          // 0..31  (M tiles over 512 rows)
  const int nt = blockIdx.y;          // 0..15  (N tiles over 256 cols)
  const int n  = nt * 16 + l16;

  v8f acc;
  const float binit = b1[n];
  #pragma unroll
  for (int j = 0; j < 8; ++j) acc[j] = binit;   // fold +b1 into C init

  const float* Arow = F  + (size_t)(mt * 16 + l16) * INF_K + 2 * half;
  const float* Bcol = W1 + (size_t)(2 * half) * HID + n;

  #pragma unroll 4
  for (int k0 = 0; k0 < INF_K; k0 += 4) {
    v2f a, b;
    const float2 av = *reinterpret_cast<const float2*>(Arow + k0); // 8B aligned
    a[0] = av.x; a[1] = av.y;
    b[0] = Bcol[(size_t)k0 * HID];
    b[1] = Bcol[(size_t)k0 * HID + HID];
    // D = A*B + C, fp32 matrix pipe
    acc = __builtin_amdgcn_wmma_f32_16x16x4_f32(
        false, a, false, b, (short)0, acc, false, false);
    if ((k0 & 63) == 0 && (k0 + 256) < INF_K)
      __builtin_prefetch(Bcol + (size_t)(k0 + 256) * HID, 0, 1); // global_prefetch_b8
  }

  float* Hrow = H + (size_t)(mt * 16 + 8 * half) * HID + nt * 16 + l16;
  #pragma unroll
  for (int j = 0; j < 8; ++j) {
    const float v = acc[j];
    Hrow[(size_t)j * HID] = v > 0.0f ? v : 0.0f;   // relu
  }
}

// ---------------------------------------------------------------------------
// K2: scores S[p] = H[p,:]@W2 + b2, then full sequential beam search.
// Single block, 128 threads; deterministic fixed-tree reductions, serial
// top-k / dedupe on thread 0 (replicates jax.lax.top_k stable-tie order and
// the tril-duplicate -> NEG_INF masking of _prune).
// ---------------------------------------------------------------------------
__global__ __launch_bounds__(128) void wd_score_beam(
    const float* __restrict__ H, const float* __restrict__ W2,
    const float* __restrict__ b2, float* __restrict__ out)
{
  __shared__ float S[NPAIR];
  __shared__ float st[2][8][33];   // beam structs (double buffered)
  __shared__ float sc[2][8][33];   // beam scores
  __shared__ float ost[8][33];     // snapshot of beam at start of each w
  __shared__ float osc[8][33];
  __shared__ float ssum[2][8];
  __shared__ float esum[128];      // expansion sums (<=128 rows)
  __shared__ int   sel[8];

  const int tid  = threadIdx.x;
  const int lane = tid & 31;
  const int wv   = tid >> 5;       // wave id 0..3

  // ---- Phase 0: pair scores (one wave per pair, fixed reduction tree) ----
  for (int p0 = 0; p0 < NPAIR; p0 += 4) {
    const int p = p0 + wv;
    if (p < NPAIR) {
      float acc = 0.0f;
      #pragma unroll
      for (int t = 0; t < 8; ++t)
        acc += H[(size_t)p * HID + lane + 32 * t] * W2[lane + 32 * t];
      for (int off = 16; off > 0; off >>= 1)
        acc += __shfl_down(acc, off, 32);
      if (lane == 0) S[p] = acc + b2[0];
    }
  }

  // ---- init beam: 1 row of zeros, 33 cols ----
  for (int c = tid; c < 33; c += 128) { st[0][0][c] = 0.0f; sc[0][0][c] = 0.0f; }
  __syncthreads();

  int k = 1;          // current beam rows
  int cur = 0;        // buffer parity

  for (int w = 1; w <= 31; ++w) {
    const int nwin = N_M - w;
    const int kold = k;
    const int cs0  = 2 * kold;
    const int soff = pair_off(w);

    // snapshot beam entering this w (the "g_struct/g_scores" local rows use it)
    for (int idx = tid; idx < kold * 33; idx += 128) {
      const int r = idx / 33, c = idx % 33;
      ost[r][c] = st[cur][r][c];
      osc[r][c] = sc[cur][r][c];
    }
    __syncthreads();

    int cur_len = w + 1;   // width of fw slice

    for (int q = 1; q <= nwin; ++q) {
      const int nxt  = cur ^ 1;
      int rows;

      // ---- expansion sums ----
      if (q == 1) {
        rows = 2 * kold;                       // concat([fw, fw]) + append
        if (tid < rows) {
          const int base = tid % kold;
          const float app = (tid < kold) ? osc[base][w + 1] : S[soff];
          float ssv = app;
          for (int c = 0; c <= w; ++c) ssv += osc[base][c];
          esum[tid] = ssv;
        }
      } else {
        rows = cs0 * k;                        // tile cross (modulo pairing)
        for (int j = tid; j < rows; j += 128) {
          const int s = j % cs0;
          const float csc = (s < kold) ? osc[s][q + w] : S[soff + q - 1];
          esum[j] = ssum[cur][j % k] + csc;
        }
      }
      const int kk = rows < 8 ? rows : 8;
      __syncthreads();

      // ---- top-k: descending, ties -> lowest index (lax.top_k) ----
      if (tid == 0) {
        unsigned used[4] = {0u, 0u, 0u, 0u};
        for (int r = 0; r < kk; ++r) {
          float best = 0.0f; int bi = -1;
          for (int j = 0; j < rows; ++j) {
            if (used[j >> 5] & (1u << (j & 31))) continue;
            if (bi < 0 || esum[j] > best) { best = esum[j]; bi = j; }
          }
          used[bi >> 5] |= 1u << (bi & 31);
          sel[r] = bi;
          ssum[nxt][r] = best;
        }
      }
      __syncthreads();

      // ---- materialize selected rows into nxt buffer ----
      const int new_len = cur_len + 1;
      for (int idx = tid; idx < kk * new_len; idx += 128) {
        const int r = idx / new_len, c = idx % new_len;
        const int j = sel[r];
        float vst, vsc;
        if (q == 1) {
          const int base = j % kold;
          if (c < cur_len) { vst = ost[base][c]; vsc = osc[base][c]; }
          else {
            const bool loc = (j < kold);
            vst = loc ? ost[base][w + 1] : 1.0f;          // new_link = q = 1
            vsc = loc ? osc[base][w + 1] : S[soff];
          }
        } else {
          const int g = j % k;
          const int s = j % cs0;
          if (c < cur_len) { vst = st[cur][g][c]; vsc = sc[cur][g][c]; }
          else {
            const bool loc = (s < kold);
            vst = loc ? ost[s][q + w] : (float)q;
            vsc = loc ? osc[s][q + w] : S[soff + q - 1];
          }
        }
        st[nxt][r][c] = vst;
        sc[nxt][r][c] = vsc;
      }
      __syncthreads();

      // ---- duplicate masking (exact row equality incl. ssum) ----
      if (tid == 0) {
        bool dup[8];
        for (int r = 0; r < kk; ++r) dup[r] = false;
        for (int r = 1; r < kk; ++r) {
          for (int r2 = 0; r2 < r; ++r2) {
            if (ssum[nxt][r] != ssum[nxt][r2]) continue;
            bool eq = true;
            for (int c = 0; c < new_len && eq; ++c)
              eq = (st[nxt][r][c] == st[nxt][r2][c]) &&
                   (sc[nxt][r][c] == sc[nxt][r2][c]);
            if (eq) { dup[r] = true; break; }
          }
        }
        for (int r = 0; r < kk; ++r)
          if (dup[r]) ssum[nxt][r] = WD_NEG_INF;
      }
      __syncthreads();

      cur = nxt;
      k = kk;
      cur_len = new_len;
    } // q
  } // w

  // ---- output: (g_struct 8x33, g_scores 8x33) flattened in return order ----
  for (int idx = tid; idx < 8 * 33; idx += 128) {
    const int r = idx / 33, c = idx % 33;
    out[idx]            = st[cur][r][c];
    out[8 * 33 + idx]   = sc[cur][r][c];
  }
}

// ---------------------------------------------------------------------------
// launcher
// ---------------------------------------------------------------------------
extern "C" void kernel_launch(void* const* d_in, const int* in_sizes, int n_in,
                              void* d_out, int out_size, void* d_ws, size_t ws_size,
                              hipStream_t stream) {
  (void)in_sizes; (void)n_in; (void)out_size; (void)ws_size;
  const float* mentions = (const float*)d_in[0];   // (32,1024)
  const float* dist_emb = (const float*)d_in[1];   // (64,64)
  const float* W1       = (const float*)d_in[2];   // (3136,256)
  const float* b1       = (const float*)d_in[3];   // (256,)
  const float* W2       = (const float*)d_in[4];   // (256,1)
  const float* b2       = (const float*)d_in[5];   // (1,)
  float* out = (float*)d_out;                      // 2 * 8 * 33 floats

  float* F = (float*)d_ws;                                         // 512x3136
  float* H = (float*)((char*)d_ws + (size_t)MPAD * INF_K * sizeof(float)); // 512x256

  wd_build_feats<<<MPAD, 256, 0, stream>>>(mentions, dist_emb, F);
  wd_gemm_relu<<<dim3(MPAD / 16, HID / 16), 32, 0, stream>>>(F, W1, b1, H);
  wd_score_beam<<<1, 128, 0, stream>>>(H, W2, b2, out);
}